// BPR_61521111547978
// MI455X (gfx1250) — compile-verified
//
#include <hip/hip_runtime.h>
#include <hip/hip_bf16.h>
#include <math.h>

#define USER_NUM 100000
#define ITEM_NUM 50000
#define FACTOR   64
#define N_EDGES  3200000
#define BATCH    16384

typedef __attribute__((ext_vector_type(2))) float v2f;
typedef __attribute__((ext_vector_type(8))) float v8f;

// ---------------------------------------------------------------------------
// uL[r,:] = src[r,:] * d[r]   (one float4 per thread; FACTOR/4 = 16 per row)
// ---------------------------------------------------------------------------
__global__ void scale_rows_kernel(const float* __restrict__ src,
                                  const float* __restrict__ d,
                                  float* __restrict__ dst, int rows) {
  int t = blockIdx.x * blockDim.x + threadIdx.x;
  int total = rows * (FACTOR / 4);
  if (t >= total) return;
  int r = t >> 4;
  float s = d[r];
  float4 x = ((const float4*)src)[t];
  float4 y;
  y.x = s * x.x; y.y = s * x.y; y.z = s * x.z; y.w = s * x.w;
  ((float4*)dst)[t] = y;
}

// ---------------------------------------------------------------------------
// dst[dst_idx[e], :] += vals[e] * src[src_idx[e], :]
// 16 lanes per edge, float4 per lane, hardware fp32 atomics into L2-resident
// accumulators (38.4 MB total -- fits MI455X's 192 MB L2).
// ---------------------------------------------------------------------------
__global__ void scatter_edges_kernel(const float* __restrict__ src,
                                     const int* __restrict__ src_idx,
                                     const int* __restrict__ dst_idx,
                                     const float* __restrict__ vals,
                                     float* __restrict__ dst, int n_edges) {
  int t = blockIdx.x * blockDim.x + threadIdx.x;
  int e = t >> 4;
  if (e >= n_edges) return;
  int l = t & 15;
  if (l == 0) {
    // run ahead on the streamed edge arrays (global_prefetch_b8)
    __builtin_prefetch(src_idx + e + 4096, 0, 1);
    __builtin_prefetch(dst_idx + e + 4096, 0, 1);
    __builtin_prefetch(vals    + e + 4096, 0, 1);
  }
  float v = vals[e];
  long  sr = src_idx[e];
  long  dr = dst_idx[e];
  float4 x = ((const float4*)(src + sr * FACTOR))[l];
  float* dp = dst + dr * FACTOR + l * 4;
  unsafeAtomicAdd(dp + 0, v * x.x);
  unsafeAtomicAdd(dp + 1, v * x.y);
  unsafeAtomicAdd(dp + 2, v * x.z);
  unsafeAtomicAdd(dp + 3, v * x.w);
}

// ---------------------------------------------------------------------------
// Batched dot products via V_WMMA_F32_16X16X4_F32.
// One wave (32 threads) per tile of 16 batch entries.
//   A (16x4): lane&15 = M,  vgpr v holds K = 2*(lane>>4) + v
//   B (4x16): lane&15 = N,  vgpr v holds K = 2*(lane>>4) + v  (B[k][n]=ie[n][k])
// => A-frag and B-frag are identical float2 loads from row (lane&15),
//    column kb + 2*(lane>>4). Diagonal of C = the wanted dot products.
// C/D layout: vgpr r: lanes 0-15 -> (M=r,   N=lane),
//                      lanes 16-31 -> (M=r+8, N=lane-16).
// Diag at lanes 0..7: c[lane]; lanes 24..31: c[lane-24].
// ---------------------------------------------------------------------------
__global__ void predict_kernel(const float* __restrict__ u0, const float* __restrict__ u1,
                               const float* __restrict__ u2, const float* __restrict__ u3,
                               const float* __restrict__ i0, const float* __restrict__ i1,
                               const float* __restrict__ i2, const float* __restrict__ i3,
                               const int* __restrict__ user, const int* __restrict__ item_i,
                               const int* __restrict__ item_j,
                               float* __restrict__ pred_i, float* __restrict__ pred_j,
                               float* __restrict__ acc) {
  __shared__ float sdi[16];
  __shared__ float sdj[16];

  const int lane = threadIdx.x;       // 0..31, wave32
  const int idx  = lane & 15;
  const int half = lane >> 4;
  const int mb   = blockIdx.x * 16;

  const long ur = user[mb + idx];
  const long ir = item_i[mb + idx];
  const long jr = item_j[mb + idx];

  const float* Up[4] = {u0, u1, u2, u3};
  const float* Ip[4] = {i0, i1, i2, i3};

  v8f ci = {};
  v8f cj = {};
  float l2acc = 0.f;

#pragma unroll
  for (int p = 0; p < 4; ++p) {
    const float* ub = Up[p] + ur * FACTOR + 2 * half;
    const float* ib = Ip[p] + ir * FACTOR + 2 * half;
    const float* jb = Ip[p] + jr * FACTOR + 2 * half;
#pragma unroll 4
    for (int kb = 0; kb < FACTOR; kb += 4) {
      v2f a  = *(const v2f*)(ub + kb);
      v2f bi = *(const v2f*)(ib + kb);
      v2f bj = *(const v2f*)(jb + kb);
      ci = __builtin_amdgcn_wmma_f32_16x16x4_f32(false, a, false, bi, (short)0, ci,
                                                 false, false);
      cj = __builtin_amdgcn_wmma_f32_16x16x4_f32(false, a, false, bj, (short)0, cj,
                                                 false, false);
      l2acc += a.x * a.x + a.y * a.y + bi.x * bi.x + bi.y * bi.y +
               bj.x * bj.x + bj.y * bj.y;
    }
  }

  // lanes (idx) and (idx+16) together cover all 256 columns of row idx
  l2acc += __shfl_xor(l2acc, 16, 32);

  // extract the diagonal of the two C tiles
  int sel = (lane < 8) ? lane : ((lane >= 24) ? (lane - 24) : -1);
  float dvi = 0.f, dvj = 0.f;
#pragma unroll
  for (int r = 0; r < 8; ++r) {
    if (r == sel) { dvi = ci[r]; dvj = cj[r]; }
  }
  int m = (lane < 16) ? lane : (lane - 16);
  if (sel >= 0) { sdi[m] = dvi; sdj[m] = dvj; }
  __syncthreads();

  if (lane < 16) {
    float pi = sdi[lane];
    float pj = sdj[lane];
    pred_i[mb + lane] = pi;
    pred_j[mb + lane] = pj;
    float x  = pi - pj;
    // log_sigmoid(x) = min(x,0) - log1p(exp(-|x|))
    float ls = fminf(x, 0.f) - log1pf(expf(-fabsf(x)));
    float l2 = 0.01f * l2acc;
    unsafeAtomicAdd(&acc[0], ls);
    unsafeAtomicAdd(&acc[1], l2);
  }
}

__global__ void zero_acc_kernel(float* __restrict__ acc) {
  if (threadIdx.x < 2) acc[threadIdx.x] = 0.f;
}

__global__ void finalize_kernel(const float* __restrict__ acc, float* __restrict__ out) {
  float loss2 = -acc[0] / (float)BATCH;
  float loss  = loss2 + acc[1] / (float)BATCH;
  out[2 * BATCH + 0] = loss;
  out[2 * BATCH + 1] = loss2;
}

// ---------------------------------------------------------------------------
extern "C" void kernel_launch(void* const* d_in, const int* in_sizes, int n_in,
                              void* d_out, int out_size, void* d_ws, size_t ws_size,
                              hipStream_t stream) {
  const float* users = (const float*)d_in[0];
  const float* items = (const float*)d_in[1];
  const float* d_i   = (const float*)d_in[2];
  const float* d_j   = (const float*)d_in[3];
  const float* ev_ui = (const float*)d_in[4];
  const float* ev_iu = (const float*)d_in[5];
  const int*   e_u   = (const int*)d_in[6];
  const int*   e_i   = (const int*)d_in[7];
  const int*   b_u   = (const int*)d_in[8];
  const int*   b_i   = (const int*)d_in[9];
  const int*   b_j   = (const int*)d_in[10];
  float* out = (float*)d_out;

  float* ws = (float*)d_ws;
  const size_t usz = (size_t)USER_NUM * FACTOR;  // 6.4M floats
  const size_t isz = (size_t)ITEM_NUM * FACTOR;  // 3.2M floats
  float* u1 = ws;
  float* u2 = u1 + usz;
  float* u3 = u2 + usz;
  float* i1 = u3 + usz;
  float* i2 = i1 + isz;
  float* i3 = i2 + isz;
  float* acc = i3 + isz;  // 2 floats

  zero_acc_kernel<<<1, 32, 0, stream>>>(acc);

  const int TB = 256;
  const int ugrid = (int)((usz / 4 + TB - 1) / TB);
  const int igrid = (int)((isz / 4 + TB - 1) / TB);
  const int egrid = (int)(((size_t)N_EDGES * 16 + TB - 1) / TB);

  // layer 1: u1 = agg_users(i0) + u0*di ; i1 = agg_items(u0) + i0*dj
  scale_rows_kernel<<<ugrid, TB, 0, stream>>>(users, d_i, u1, USER_NUM);
  scale_rows_kernel<<<igrid, TB, 0, stream>>>(items, d_j, i1, ITEM_NUM);
  scatter_edges_kernel<<<egrid, TB, 0, stream>>>(items, e_i, e_u, ev_ui, u1, N_EDGES);
  scatter_edges_kernel<<<egrid, TB, 0, stream>>>(users, e_u, e_i, ev_iu, i1, N_EDGES);

  // layer 2
  scale_rows_kernel<<<ugrid, TB, 0, stream>>>(u1, d_i, u2, USER_NUM);
  scale_rows_kernel<<<igrid, TB, 0, stream>>>(i1, d_j, i2, ITEM_NUM);
  scatter_edges_kernel<<<egrid, TB, 0, stream>>>(i1, e_i, e_u, ev_ui, u2, N_EDGES);
  scatter_edges_kernel<<<egrid, TB, 0, stream>>>(u1, e_u, e_i, ev_iu, i2, N_EDGES);

  // layer 3
  scale_rows_kernel<<<ugrid, TB, 0, stream>>>(u2, d_i, u3, USER_NUM);
  scale_rows_kernel<<<igrid, TB, 0, stream>>>(i2, d_j, i3, ITEM_NUM);
  scatter_edges_kernel<<<egrid, TB, 0, stream>>>(i2, e_i, e_u, ev_ui, u3, N_EDGES);
  scatter_edges_kernel<<<egrid, TB, 0, stream>>>(u2, e_u, e_i, ev_iu, i3, N_EDGES);

  // predictions + loss terms (WMMA diagonal trick), then scalars
  predict_kernel<<<BATCH / 16, 32, 0, stream>>>(users, u1, u2, u3,
                                                items, i1, i2, i3,
                                                b_u, b_i, b_j,
                                                out, out + BATCH, acc);
  finalize_kernel<<<1, 1, 0, stream>>>(acc, out);
}